// SubpixDenseUP_11132555231953
// MI455X (gfx1250) — compile-verified
//
#include <hip/hip_runtime.h>
#include <stdint.h>

// Native vector types (clang builtins want real vector types, not HIP structs).
typedef float v4f __attribute__((ext_vector_type(4)));
typedef int   v4i __attribute__((ext_vector_type(4)));

// ---- CDNA5 async-to-LDS path probes -------------------------------------
#if defined(__HIP_DEVICE_COMPILE__) && __has_builtin(__builtin_amdgcn_global_load_async_to_lds_b128)
#define ASYNC_LDS_BUILTIN 1
#else
#define ASYNC_LDS_BUILTIN 0
#endif

#if defined(__HIP_DEVICE_COMPILE__) && __has_builtin(__builtin_amdgcn_s_wait_asynccnt)
#define WAIT_ASYNC_BUILTIN 1
#else
#define WAIT_ASYNC_BUILTIN 0
#endif

// Address-space-qualified v4i pointers for the async builtin.
// NOTE: param 0 is a NON-const AS(1) v4i* (clang prints AS(1) as "__device__").
typedef __attribute__((address_space(1))) v4i* gv4i_ptr;  // global
typedef __attribute__((address_space(3))) v4i* lv4i_ptr;  // LDS

__device__ __forceinline__ void async_load_b128_to_lds(const void* gptr, uint32_t lds_byte_off) {
#if ASYNC_LDS_BUILTIN
    __builtin_amdgcn_global_load_async_to_lds_b128(
        (gv4i_ptr)(uintptr_t)gptr,
        (lv4i_ptr)lds_byte_off,
        /*offset=*/0, /*cpol=*/0);
#else
    uint64_t ga = (uint64_t)(uintptr_t)gptr;
    uint32_t la = lds_byte_off;
    // VGLOBAL async load: VDST = LDS byte address VGPR, VADDR = 64-bit addr pair
    asm volatile("global_load_async_to_lds_b128 %0, %1, off"
                 :: "v"(la), "v"(ga) : "memory");
#endif
}

__device__ __forceinline__ void wait_asynccnt0() {
#if WAIT_ASYNC_BUILTIN
    __builtin_amdgcn_s_wait_asynccnt(0);
#else
    asm volatile("s_wait_asynccnt 0" ::: "memory");
#endif
}

// out[n, 2i+p, 2j+q, c] = x[n, i, j, 4c + 2q + p]
// x: (4, 256, 256, 256) f32 ; out: (4, 512, 512, 64) f32.
// One thread handles one 64-byte input chunk (input channels 16t..16t+15 of
// one pixel) = 4 float4 groups; group k's lanes .x/.y/.z/.w are the four
// (p,q) sub-pixel outputs at channel c = 4t+k.
__global__ __launch_bounds__(256) void subpix_up_async(const float* __restrict__ x,
                                                       float* __restrict__ out) {
    __shared__ float smem[4096];            // 256 threads * 64B = 16 KB

    const int tid = threadIdx.x;
    const int g   = blockIdx.x * 256 + tid; // global 64B-chunk index
    const size_t in_base = (size_t)g * 16;  // floats

    // Stage this thread's 64B input slice into LDS via async DMA (ASYNCcnt).
    const char* gp = (const char*)(x + in_base);
    const uint32_t lbase = (uint32_t)(uintptr_t)smem + (uint32_t)tid * 64u;
#pragma unroll
    for (int k = 0; k < 4; ++k)
        async_load_b128_to_lds(gp + k * 16, lbase + (uint32_t)(k * 16));

    wait_asynccnt0();
    __syncthreads();

    const v4f* s4 = reinterpret_cast<const v4f*>(&smem[tid * 16]);
    v4f v0 = s4[0], v1 = s4[1], v2 = s4[2], v3 = s4[3];

    // Decode pixel coordinates.
    const int t     = g & 15;          // 16 threads per pixel
    const int pixel = g >> 4;
    const int j = pixel & 255;
    const int i = (pixel >> 8) & 255;
    const int n = pixel >> 16;

    // Flat output float index of (n, 2i, 2j, 4t).
    const size_t row0 = (((size_t)(n * 512 + 2 * i) * 512) + (size_t)(2 * j)) * 64
                        + (size_t)(4 * t);
    v4f* o = reinterpret_cast<v4f*>(out);
    const size_t o0 = row0 >> 2;       // in v4f units (row0 is a multiple of 4)

    // component m = 2q + p of group k maps to out(p,q) channel c = 4t+k
    v4f pq00 = (v4f){v0.x, v1.x, v2.x, v3.x}; // p=0,q=0
    v4f pq10 = (v4f){v0.y, v1.y, v2.y, v3.y}; // p=1,q=0
    v4f pq01 = (v4f){v0.z, v1.z, v2.z, v3.z}; // p=0,q=1
    v4f pq11 = (v4f){v0.w, v1.w, v2.w, v3.w}; // p=1,q=1

    // Streaming output: non-temporal 128-bit stores.
    __builtin_nontemporal_store(pq00, &o[o0]);                 // row 2i,   col 2j
    __builtin_nontemporal_store(pq01, &o[o0 + 16]);            // row 2i,   col 2j+1 (+64 floats)
    __builtin_nontemporal_store(pq10, &o[o0 + 8192]);          // row 2i+1, col 2j   (+512*64 floats)
    __builtin_nontemporal_store(pq11, &o[o0 + 8192 + 16]);     // row 2i+1, col 2j+1
}

extern "C" void kernel_launch(void* const* d_in, const int* in_sizes, int n_in,
                              void* d_out, int out_size, void* d_ws, size_t ws_size,
                              hipStream_t stream) {
    (void)in_sizes; (void)n_in; (void)d_ws; (void)ws_size;
    const float* x = (const float*)d_in[0];
    float* out = (float*)d_out;

    // out_size = 4*512*512*64 = 67,108,864 floats; one thread per 16 floats.
    const int total_chunks = out_size / 16;          // 4,194,304
    const int blocks = total_chunks / 256;           // 16,384
    subpix_up_async<<<blocks, 256, 0, stream>>>(x, out);
}